// GAT_89215060673333
// MI455X (gfx1250) — compile-verified
//
#include <hip/hip_runtime.h>
#include <hip/hip_bf16.h>

#define N_NODESK 50000
#define N_EDGESK 1600000
#define EE (N_EDGESK + N_NODESK)   // edges + self loops = 1,650,000
#define IN_CH 128
#define HID 32
#define HEADS 2
#define C1 (HEADS * HID)           // 64
#define OUT_CH 16
#define NEG_SLOPE 0.2f

typedef float v2f __attribute__((ext_vector_type(2)));
typedef float v8f __attribute__((ext_vector_type(8)));

__device__ __forceinline__ void atomic_max_f32(float* addr, float v) {
  // classic monotonic-bits trick; buffer must be initialized to -inf
  if (v >= 0.0f) (void)atomicMax((int*)addr, __float_as_int(v));
  else           (void)atomicMin((unsigned int*)addr, __float_as_uint(v));
}

__device__ __forceinline__ float lrelu(float x) {
  return x > 0.0f ? x : NEG_SLOPE * x;
}

__device__ __forceinline__ void edge_sd(const int* __restrict__ ei, int i,
                                        int& s, int& d) {
  if (i < N_EDGESK) { s = ei[i]; d = ei[N_EDGESK + i]; }
  else              { s = i - N_EDGESK; d = s; }       // self loop
}

// ---------------- utility ----------------
__global__ void fill_f32(float* __restrict__ p, float v, int n) {
  int i = blockIdx.x * blockDim.x + threadIdx.x;
  if (i < n) p[i] = v;
}

// ---------------- GEMM1: h1[N,64] = x[N,128] @ W1[128,64] (f32 WMMA) ----------------
__global__ __launch_bounds__(128) void gemm1_wmma(
    const float* __restrict__ x, const float* __restrict__ W1,
    float* __restrict__ h1) {
  __shared__ float xs[16 * IN_CH];              // 8 KB A tile (16 rows x 128)
  const int m0 = blockIdx.x * 16;               // 3125 blocks, exact tiling
  for (int e = threadIdx.x; e < 16 * IN_CH; e += 128)
    xs[e] = x[m0 * IN_CH + e];                  // contiguous cooperative load
  __syncthreads();

  const int wave = threadIdx.x >> 5;            // 4 waves -> 4 n-tiles of 16
  const int lane = threadIdx.x & 31;
  const int half = lane >> 4;                   // A/B K-pair select
  const int l    = lane & 15;
  const int n0   = wave * 16;

  v8f c = {};
  #pragma unroll 4
  for (int k = 0; k < IN_CH; k += 4) {
    const int kk = k + 2 * half;
    float2 av = *(const float2*)(&xs[l * IN_CH + kk]);   // A: row l, K=kk,kk+1
    v2f a; a.x = av.x; a.y = av.y;
    v2f b;                                               // B: col l, K=kk,kk+1
    b.x = W1[(kk + 0) * C1 + n0 + l];
    b.y = W1[(kk + 1) * C1 + n0 + l];
    c = __builtin_amdgcn_wmma_f32_16x16x4_f32(false, a, false, b, (short)0, c,
                                              false, false);
  }
  #pragma unroll
  for (int i = 0; i < 8; ++i)                   // C/D: VGPR i -> rows i / i+8
    h1[(m0 + i + 8 * half) * C1 + n0 + l] = c[i];
}

// ---------------- GEMM2: t2[N,16] = h2[N,64] @ W2[64,16] (f32 WMMA) ----------------
__global__ __launch_bounds__(128) void gemm2_wmma(
    const float* __restrict__ h2, const float* __restrict__ W2,
    float* __restrict__ t2) {
  const int wave = threadIdx.x >> 5;
  const int mt   = blockIdx.x * 4 + wave;       // one 16-row tile per wave
  if (mt >= N_NODESK / 16) return;              // wave-uniform: EXEC stays all-1
  const int lane = threadIdx.x & 31;
  const int half = lane >> 4;
  const int l    = lane & 15;
  const int m0   = mt * 16;

  v8f c = {};
  #pragma unroll
  for (int k = 0; k < C1; k += 4) {
    const int kk = k + 2 * half;
    float2 av = *(const float2*)(&h2[(m0 + l) * C1 + kk]);
    v2f a; a.x = av.x; a.y = av.y;
    v2f b;
    b.x = W2[(kk + 0) * OUT_CH + l];
    b.y = W2[(kk + 1) * OUT_CH + l];
    c = __builtin_amdgcn_wmma_f32_16x16x4_f32(false, a, false, b, (short)0, c,
                                              false, false);
  }
  #pragma unroll
  for (int i = 0; i < 8; ++i)
    t2[(m0 + i + 8 * half) * OUT_CH + l] = c[i];
}

// ---------------- per-node attention halves ----------------
__global__ void node_att1(const float* __restrict__ h1,
                          const float* __restrict__ a_src,
                          const float* __restrict__ a_dst,
                          float* __restrict__ als, float* __restrict__ ald) {
  int n = blockIdx.x * blockDim.x + threadIdx.x;
  if (n >= N_NODESK) return;
  #pragma unroll
  for (int hd = 0; hd < HEADS; ++hd) {
    float s = 0.f, d = 0.f;
    for (int cc = 0; cc < HID; ++cc) {
      float hv = h1[n * C1 + hd * HID + cc];
      s = fmaf(hv, a_src[hd * HID + cc], s);
      d = fmaf(hv, a_dst[hd * HID + cc], d);
    }
    als[n * HEADS + hd] = s;
    ald[n * HEADS + hd] = d;
  }
}

__global__ void node_att2(const float* __restrict__ t2,
                          const float* __restrict__ a_src,
                          const float* __restrict__ a_dst,
                          float* __restrict__ als, float* __restrict__ ald) {
  int n = blockIdx.x * blockDim.x + threadIdx.x;
  if (n >= N_NODESK) return;
  float s = 0.f, d = 0.f;
  for (int cc = 0; cc < OUT_CH; ++cc) {
    float hv = t2[n * OUT_CH + cc];
    s = fmaf(hv, a_src[cc], s);
    d = fmaf(hv, a_dst[cc], d);
  }
  als[n] = s;
  ald[n] = d;
}

// ---------------- layer-1 edge passes ----------------
__global__ void edge_max1(const int* __restrict__ ei,
                          const float* __restrict__ als,
                          const float* __restrict__ ald,
                          float* __restrict__ m1) {
  int i = blockIdx.x * blockDim.x + threadIdx.x;
  if (i >= EE) return;
  int s, d; edge_sd(ei, i, s, d);
  #pragma unroll
  for (int hd = 0; hd < HEADS; ++hd) {
    float e = lrelu(als[s * HEADS + hd] + ald[d * HEADS + hd]);
    atomic_max_f32(&m1[d * HEADS + hd], e);
  }
}

__global__ void edge_exp1(const int* __restrict__ ei,
                          const float* __restrict__ als,
                          const float* __restrict__ ald,
                          const float* __restrict__ m1,
                          float* __restrict__ ex1, float* __restrict__ den1) {
  int i = blockIdx.x * blockDim.x + threadIdx.x;
  if (i >= EE) return;
  int s, d; edge_sd(ei, i, s, d);
  #pragma unroll
  for (int hd = 0; hd < HEADS; ++hd) {
    float e  = lrelu(als[s * HEADS + hd] + ald[d * HEADS + hd]);
    float ex = expf(e - m1[d * HEADS + hd]);
    ex1[i * HEADS + hd] = ex;
    atomicAdd(&den1[d * HEADS + hd], ex);
  }
}

__global__ void edge_agg1(const int* __restrict__ ei,
                          const float* __restrict__ h1,
                          const float* __restrict__ ex1,
                          const float* __restrict__ den1,
                          float* __restrict__ agg) {
  int i = blockIdx.x * blockDim.x + threadIdx.x;   // over EE*64
  if (i >= EE * C1) return;
  int edge = i >> 6;
  int ch   = i & (C1 - 1);
  int hd   = ch >> 5;
  int s, d; edge_sd(ei, edge, s, d);
  float alpha = ex1[edge * HEADS + hd] / (den1[d * HEADS + hd] + 1e-16f);
  atomicAdd(&agg[d * C1 + ch], alpha * h1[s * C1 + ch]);
}

__global__ void bias_relu1(float* __restrict__ h2, const float* __restrict__ b1) {
  int i = blockIdx.x * blockDim.x + threadIdx.x;
  if (i >= N_NODESK * C1) return;
  float v = h2[i] + b1[i & (C1 - 1)];
  h2[i] = v > 0.f ? v : 0.f;
}

// ---------------- layer-2 edge passes ----------------
__global__ void edge_max2(const int* __restrict__ ei,
                          const float* __restrict__ als,
                          const float* __restrict__ ald,
                          float* __restrict__ m2) {
  int i = blockIdx.x * blockDim.x + threadIdx.x;
  if (i >= EE) return;
  int s, d; edge_sd(ei, i, s, d);
  float e = lrelu(als[s] + ald[d]);
  atomic_max_f32(&m2[d], e);
}

__global__ void edge_exp2(const int* __restrict__ ei,
                          const float* __restrict__ als,
                          const float* __restrict__ ald,
                          const float* __restrict__ m2,
                          float* __restrict__ ex2, float* __restrict__ den2) {
  int i = blockIdx.x * blockDim.x + threadIdx.x;
  if (i >= EE) return;
  int s, d; edge_sd(ei, i, s, d);
  float e  = lrelu(als[s] + ald[d]);
  float ex = expf(e - m2[d]);
  ex2[i] = ex;
  atomicAdd(&den2[d], ex);
}

__global__ void edge_agg2(const int* __restrict__ ei,
                          const float* __restrict__ t2,
                          const float* __restrict__ ex2,
                          const float* __restrict__ den2,
                          float* __restrict__ out) {
  int i = blockIdx.x * blockDim.x + threadIdx.x;   // over EE*16
  if (i >= EE * OUT_CH) return;
  int edge = i >> 4;
  int ch   = i & (OUT_CH - 1);
  int s, d; edge_sd(ei, edge, s, d);
  float alpha = ex2[edge] / (den2[d] + 1e-16f);
  atomicAdd(&out[d * OUT_CH + ch], alpha * t2[s * OUT_CH + ch]);
}

__global__ void bias2(float* __restrict__ out, const float* __restrict__ b2) {
  int i = blockIdx.x * blockDim.x + threadIdx.x;
  if (i >= N_NODESK * OUT_CH) return;
  out[i] += b2[i & (OUT_CH - 1)];
}

// ---------------- launcher ----------------
extern "C" void kernel_launch(void* const* d_in, const int* in_sizes, int n_in,
                              void* d_out, int out_size, void* d_ws, size_t ws_size,
                              hipStream_t stream) {
  const float* x   = (const float*)d_in[0];
  const int*   ei  = (const int*)  d_in[1];
  const float* W1  = (const float*)d_in[2];
  const float* as1 = (const float*)d_in[3];
  const float* ad1 = (const float*)d_in[4];
  const float* b1  = (const float*)d_in[5];
  const float* W2  = (const float*)d_in[6];
  const float* as2 = (const float*)d_in[7];
  const float* ad2 = (const float*)d_in[8];
  const float* b2  = (const float*)d_in[9];
  float* out = (float*)d_out;

  // workspace carve-up (floats)
  float* ws   = (float*)d_ws;
  float* h1   = ws;                         // 50000*64   = 3,200,000
  float* als1 = h1   + N_NODESK * C1;       // 100,000
  float* ald1 = als1 + N_NODESK * HEADS;    // 100,000
  float* m1   = ald1 + N_NODESK * HEADS;    // 100,000
  float* den1 = m1   + N_NODESK * HEADS;    // 100,000
  float* ex1  = den1 + N_NODESK * HEADS;    // 3,300,000
  float* h2   = ex1  + EE * HEADS;          // 3,200,000 (agg1 -> relu -> layer2 in)
  float* t2   = h2   + N_NODESK * C1;       // 800,000
  float* als2 = t2   + N_NODESK * OUT_CH;   // 50,000
  float* ald2 = als2 + N_NODESK;            // 50,000
  float* m2   = ald2 + N_NODESK;            // 50,000
  float* den2 = m2   + N_NODESK;            // 50,000
  float* ex2  = den2 + N_NODESK;            // 1,650,000

  const int T = 256;
  #define GRID(n) (((n) + T - 1) / T)
  #define FILL(p, v, n) fill_f32<<<GRID(n), T, 0, stream>>>((p), (v), (n))

  // init accumulators
  FILL(m1,   -INFINITY, N_NODESK * HEADS);
  FILL(den1, 0.0f,      N_NODESK * HEADS);
  FILL(h2,   0.0f,      N_NODESK * C1);
  FILL(m2,   -INFINITY, N_NODESK);
  FILL(den2, 0.0f,      N_NODESK);
  FILL(out,  0.0f,      N_NODESK * OUT_CH);

  // ----- layer 1 -----
  gemm1_wmma<<<N_NODESK / 16, 128, 0, stream>>>(x, W1, h1);
  node_att1<<<GRID(N_NODESK), T, 0, stream>>>(h1, as1, ad1, als1, ald1);
  edge_max1<<<GRID(EE), T, 0, stream>>>(ei, als1, ald1, m1);
  edge_exp1<<<GRID(EE), T, 0, stream>>>(ei, als1, ald1, m1, ex1, den1);
  edge_agg1<<<GRID(EE * C1), T, 0, stream>>>(ei, h1, ex1, den1, h2);
  bias_relu1<<<GRID(N_NODESK * C1), T, 0, stream>>>(h2, b1);

  // ----- layer 2 -----
  gemm2_wmma<<<(N_NODESK / 16 + 3) / 4, 128, 0, stream>>>(h2, W2, t2);
  node_att2<<<GRID(N_NODESK), T, 0, stream>>>(t2, as2, ad2, als2, ald2);
  edge_max2<<<GRID(EE), T, 0, stream>>>(ei, als2, ald2, m2);
  edge_exp2<<<GRID(EE), T, 0, stream>>>(ei, als2, ald2, m2, ex2, den2);
  edge_agg2<<<GRID(EE * OUT_CH), T, 0, stream>>>(ei, t2, ex2, den2, out);
  bias2<<<GRID(N_NODESK * OUT_CH), T, 0, stream>>>(out, b2);

  #undef FILL
  #undef GRID
}